// GAT_47949014893119
// MI455X (gfx1250) — compile-verified
//
#include <hip/hip_runtime.h>
#include <hip/hip_bf16.h>
#include <math.h>

typedef __attribute__((ext_vector_type(2))) float v2f;
typedef __attribute__((ext_vector_type(8))) float v8f;

#define GAT_N      50000
#define GAT_E      800000
#define GAT_ETOT   (GAT_E + GAT_N)   // edges + self loops
#define IN_CH      128
#define HID        32
#define HEADS      8
#define OUT_CH     64
#define HC1        (HEADS * HID)     // 256
#define NEG_SLOPE  0.2f

// ---------- helpers ----------
static __device__ __forceinline__ unsigned f2ord(float f) {
    unsigned u = __float_as_uint(f);
    return (u & 0x80000000u) ? ~u : (u | 0x80000000u);
}
static __device__ __forceinline__ float ord2f(unsigned u) {
    return (u & 0x80000000u) ? __uint_as_float(u & 0x7FFFFFFFu)
                             : __uint_as_float(~u);
}
#define ORD_NEG_INF 0x007FFFFFu   // f2ord(-inf)

__global__ void gat_fill_f32(float* p, float v, int n) {
    int i = blockIdx.x * blockDim.x + threadIdx.x;
    if (i < n) p[i] = v;
}
__global__ void gat_fill_u32(unsigned* p, unsigned v, int n) {
    int i = blockIdx.x * blockDim.x + threadIdx.x;
    if (i < n) p[i] = v;
}

// ---------- fp32 WMMA GEMM: C[M,Nc] = A[M,K] @ B[K,Nc] ----------
// One wave computes one 16x16 tile; 8 waves / block. M % 16 == 0, Nc % 16 == 0, K % 4 == 0.
// A-frag (16x4 f32): lane L (r=L&15, half=L>>4): {A[r, k0+2h], A[r, k0+2h+1]}
// B-frag (4x16 f32): lane L: {B[k0+2h, c], B[k0+2h+1, c]}, c = tile_n*16 + r
// D (16x16 f32): vgpr j -> row = tile_m*16 + 8*half + j, col = tile_n*16 + r
__global__ void gat_gemm_wmma(const float* __restrict__ A, const float* __restrict__ B,
                              float* __restrict__ C, int M, int K, int Nc) {
    int wave = blockIdx.x * (blockDim.x >> 5) + (threadIdx.x >> 5);
    int ntn = Nc >> 4;
    int total = (M >> 4) * ntn;
    if (wave >= total) return;                 // wave-uniform branch
    int tm = wave / ntn;
    int tn = wave - tm * ntn;
    int lane = threadIdx.x & 31;
    int half = lane >> 4;
    int r    = lane & 15;

    const float* Arow = A + (long)(tm * 16 + r) * K + 2 * half;
    const float* Bcol = B + (tn * 16 + r) + (long)(2 * half) * Nc;

    v8f acc = {};
    for (int k0 = 0; k0 < K; k0 += 4) {
        v2f a = *(const v2f*)(Arow + k0);      // 8B-aligned: K%4==0, 2*half even
        v2f b;
        b.x = Bcol[(long)k0 * Nc];
        b.y = Bcol[(long)k0 * Nc + Nc];
        acc = __builtin_amdgcn_wmma_f32_16x16x4_f32(
                  false, a, false, b, (short)0, acc, false, false);
    }

    float* Crow = C + (long)(tm * 16 + 8 * half) * Nc + (tn * 16 + r);
#pragma unroll
    for (int j = 0; j < 8; ++j)
        Crow[(long)j * Nc] = acc[j];
}

// ---------- per-(node,head) attention logits ----------
__global__ void gat_alpha(const float* __restrict__ h,
                          const float* __restrict__ att_s,
                          const float* __restrict__ att_d,
                          float* __restrict__ as_, float* __restrict__ ad_,
                          int n, int H, int Cc) {
    int i = blockIdx.x * blockDim.x + threadIdx.x;  // i = node*H + head
    if (i >= n * H) return;
    int node = i / H;
    int head = i - node * H;
    const float* hp = h + (long)node * H * Cc + head * Cc;
    const float* sp = att_s + head * Cc;
    const float* dp = att_d + head * Cc;
    float ss = 0.f, dd = 0.f;
    for (int c = 0; c < Cc; ++c) {
        float v = hp[c];
        ss += v * sp[c];
        dd += v * dp[c];
    }
    as_[i] = ss;
    ad_[i] = dd;
}

// ---------- edge: alpha = lrelu(as[src]+ad[dst]); segment max ----------
__global__ void gat_edge_alpha_max(const long long* __restrict__ ei,
                                   const float* __restrict__ as_,
                                   const float* __restrict__ ad_,
                                   float* __restrict__ albuf,
                                   unsigned* __restrict__ mx,
                                   int etot, int ne, int H) {
    int i = blockIdx.x * blockDim.x + threadIdx.x;
    if (i >= etot * H) return;
    int e = i / H;
    int h = i - e * H;
    int s, d;
    if (e < ne) { s = (int)ei[e]; d = (int)ei[ne + e]; }
    else        { s = d = e - ne; }
    float a = as_[(long)s * H + h] + ad_[(long)d * H + h];
    a = a > 0.f ? a : NEG_SLOPE * a;
    albuf[i] = a;
    atomicMax(mx + (long)d * H + h, f2ord(a));
}

// ---------- edge: e = exp(alpha - max[dst]); segment sum ----------
__global__ void gat_edge_exp_sum(const long long* __restrict__ ei,
                                 float* __restrict__ albuf,
                                 const unsigned* __restrict__ mx,
                                 float* __restrict__ denom,
                                 int etot, int ne, int H) {
    int i = blockIdx.x * blockDim.x + threadIdx.x;
    if (i >= etot * H) return;
    int e = i / H;
    int h = i - e * H;
    int d = (e < ne) ? (int)ei[ne + e] : (e - ne);
    float ex = expf(albuf[i] - ord2f(mx[(long)d * H + h]));
    albuf[i] = ex;
    atomicAdd(denom + (long)d * H + h, ex);
}

// ---------- edge: out[dst] += h[src] * (e/denom[dst]) ----------
// one thread per (edge, channel); HC = H*C = 1<<lHC, C = 1<<lC
__global__ void gat_msg(const long long* __restrict__ ei,
                        const float* __restrict__ hfeat,
                        const float* __restrict__ ebuf,
                        const float* __restrict__ denom,
                        float* __restrict__ out,
                        int etot, int ne, int H, int lC, int lHC) {
    long idx = (long)blockIdx.x * blockDim.x + threadIdx.x;
    if (idx >= ((long)etot << lHC)) return;
    int e = (int)(idx >> lHC);
    int c = (int)(idx & ((1 << lHC) - 1));
    int h = c >> lC;
    int s, d;
    if (e < ne) { s = (int)ei[e]; d = (int)ei[ne + e]; }
    else        { s = d = e - ne; }
    const float* srow = hfeat + ((long)s << lHC);
    if (c == 0) __builtin_prefetch(srow, 0, 0);
    float coef = ebuf[(long)e * H + h] / denom[(long)d * H + h];
    atomicAdd(out + ((long)d << lHC) + c, srow[c] * coef);
}

// ---------- elementwise: x = elu(x + b[col]) ----------
__global__ void gat_elu_bias(float* __restrict__ x, const float* __restrict__ b,
                             int n, int cols) {
    int i = blockIdx.x * blockDim.x + threadIdx.x;
    if (i >= n) return;
    float v = x[i] + b[i % cols];
    x[i] = v > 0.f ? v : (expf(v) - 1.f);
}

// ---------- final: out = agg + b[col] ----------
__global__ void gat_add_bias(const float* __restrict__ agg, const float* __restrict__ b,
                             float* __restrict__ out, int n, int cols) {
    int i = blockIdx.x * blockDim.x + threadIdx.x;
    if (i >= n) return;
    out[i] = agg[i] + b[i % cols];
}

static inline int cdiv(long a, long b) { return (int)((a + b - 1) / b); }

extern "C" void kernel_launch(void* const* d_in, const int* in_sizes, int n_in,
                              void* d_out, int out_size, void* d_ws, size_t ws_size,
                              hipStream_t stream) {
    const float*     x        = (const float*)d_in[0];
    const long long* ei       = (const long long*)d_in[1];   // [2, E] int64
    const float*     W1       = (const float*)d_in[2];       // [128, 256]
    const float*     att_src1 = (const float*)d_in[3];       // [8, 32]
    const float*     att_dst1 = (const float*)d_in[4];
    const float*     b1       = (const float*)d_in[5];       // [256]
    const float*     W2       = (const float*)d_in[6];       // [256, 64]
    const float*     att_src2 = (const float*)d_in[7];       // [1, 64]
    const float*     att_dst2 = (const float*)d_in[8];
    const float*     b2       = (const float*)d_in[9];       // [64]
    float* out = (float*)d_out;

    // ---- workspace carve (bytes, 256B-aligned chunks) ----
    char* w = (char*)d_ws;
    size_t off = 0;
    auto take = [&](size_t bytes) {
        void* p = w + off;
        off += (bytes + 255) & ~size_t(255);
        return p;
    };
    float*    h1     = (float*)take((size_t)GAT_N * HC1 * 4);     // x@W1
    float*    out1   = (float*)take((size_t)GAT_N * HC1 * 4);     // layer1 aggregate / layer2 input
    float*    h2     = (float*)take((size_t)GAT_N * OUT_CH * 4);  // out1@W2
    float*    agg2   = (float*)take((size_t)GAT_N * OUT_CH * 4);
    float*    as1    = (float*)take((size_t)GAT_N * HEADS * 4);
    float*    ad1    = (float*)take((size_t)GAT_N * HEADS * 4);
    unsigned* m1     = (unsigned*)take((size_t)GAT_N * HEADS * 4);
    float*    den1   = (float*)take((size_t)GAT_N * HEADS * 4);
    float*    eb1    = (float*)take((size_t)GAT_ETOT * HEADS * 4);
    float*    as2    = (float*)take((size_t)GAT_N * 4);
    float*    ad2    = (float*)take((size_t)GAT_N * 4);
    unsigned* m2     = (unsigned*)take((size_t)GAT_N * 4);
    float*    den2   = (float*)take((size_t)GAT_N * 4);
    float*    eb2    = (float*)take((size_t)GAT_ETOT * 4);
    (void)ws_size; (void)in_sizes; (void)n_in; (void)out_size;

    const int T = 256;

    // ================= Layer 1 =================
    // h1 = x @ W1   (M=50000, K=128, Nc=256) -> 50000 tiles, 8 waves/block
    {
        int waves = (GAT_N / 16) * (HC1 / 16);
        gat_gemm_wmma<<<cdiv(waves, 8), 256, 0, stream>>>(x, W1, h1, GAT_N, IN_CH, HC1);
    }
    gat_alpha<<<cdiv((long)GAT_N * HEADS, T), T, 0, stream>>>(h1, att_src1, att_dst1,
                                                             as1, ad1, GAT_N, HEADS, HID);
    gat_fill_u32<<<cdiv((long)GAT_N * HEADS, T), T, 0, stream>>>(m1, ORD_NEG_INF, GAT_N * HEADS);
    gat_fill_f32<<<cdiv((long)GAT_N * HEADS, T), T, 0, stream>>>(den1, 0.f, GAT_N * HEADS);
    gat_fill_f32<<<cdiv((long)GAT_N * HC1, T), T, 0, stream>>>(out1, 0.f, GAT_N * HC1);

    gat_edge_alpha_max<<<cdiv((long)GAT_ETOT * HEADS, T), T, 0, stream>>>(
        ei, as1, ad1, eb1, m1, GAT_ETOT, GAT_E, HEADS);
    gat_edge_exp_sum<<<cdiv((long)GAT_ETOT * HEADS, T), T, 0, stream>>>(
        ei, eb1, m1, den1, GAT_ETOT, GAT_E, HEADS);
    gat_msg<<<cdiv((long)GAT_ETOT * HC1, T), T, 0, stream>>>(
        ei, h1, eb1, den1, out1, GAT_ETOT, GAT_E, HEADS, /*lC=*/5, /*lHC=*/8);
    gat_elu_bias<<<cdiv((long)GAT_N * HC1, T), T, 0, stream>>>(out1, b1, GAT_N * HC1, HC1);

    // ================= Layer 2 =================
    // h2 = out1 @ W2  (M=50000, K=256, Nc=64)
    {
        int waves = (GAT_N / 16) * (OUT_CH / 16);
        gat_gemm_wmma<<<cdiv(waves, 8), 256, 0, stream>>>(out1, W2, h2, GAT_N, HC1, OUT_CH);
    }
    gat_alpha<<<cdiv((long)GAT_N, T), T, 0, stream>>>(h2, att_src2, att_dst2,
                                                     as2, ad2, GAT_N, 1, OUT_CH);
    gat_fill_u32<<<cdiv((long)GAT_N, T), T, 0, stream>>>(m2, ORD_NEG_INF, GAT_N);
    gat_fill_f32<<<cdiv((long)GAT_N, T), T, 0, stream>>>(den2, 0.f, GAT_N);
    gat_fill_f32<<<cdiv((long)GAT_N * OUT_CH, T), T, 0, stream>>>(agg2, 0.f, GAT_N * OUT_CH);

    gat_edge_alpha_max<<<cdiv((long)GAT_ETOT, T), T, 0, stream>>>(
        ei, as2, ad2, eb2, m2, GAT_ETOT, GAT_E, 1);
    gat_edge_exp_sum<<<cdiv((long)GAT_ETOT, T), T, 0, stream>>>(
        ei, eb2, m2, den2, GAT_ETOT, GAT_E, 1);
    gat_msg<<<cdiv((long)GAT_ETOT * OUT_CH, T), T, 0, stream>>>(
        ei, h2, eb2, den2, agg2, GAT_ETOT, GAT_E, 1, /*lC=*/6, /*lHC=*/6);

    gat_add_bias<<<cdiv((long)GAT_N * OUT_CH, T), T, 0, stream>>>(
        agg2, b2, out, GAT_N * OUT_CH, OUT_CH);
}